// AGNNNet_70050916598071
// MI455X (gfx1250) — compile-verified
//
#include <hip/hip_runtime.h>
#include <hip/hip_bf16.h>

typedef __attribute__((ext_vector_type(2))) float v2f;
typedef __attribute__((ext_vector_type(8))) float v8f;
typedef __attribute__((ext_vector_type(4))) unsigned int v4u;
typedef __attribute__((ext_vector_type(8))) int v8i_;
typedef __attribute__((ext_vector_type(4))) int v4i_;

#define D 16
#define AGG_CHUNK 1024

#if defined(__has_builtin)
#if __has_builtin(__builtin_amdgcn_tensor_load_to_lds) && \
    __has_builtin(__builtin_amdgcn_s_wait_tensorcnt)
#define USE_TDM 1
#endif
#endif

// ---------------------------------------------------------------------------
// h[n][d] = relu(x[n] * w[d] + b[d])      (Linear(1,16) + ReLU)
// ---------------------------------------------------------------------------
__global__ void lin_relu_kernel(const float* __restrict__ x,
                                const float* __restrict__ w,
                                const float* __restrict__ b,
                                float* __restrict__ h, int n) {
    int i = blockIdx.x * blockDim.x + threadIdx.x;
    if (i >= n) return;
    float v = x[i];
    float4* out = (float4*)(h + (size_t)i * D);
    const float4* w4 = (const float4*)w;
    const float4* b4 = (const float4*)b;
#pragma unroll
    for (int c = 0; c < 4; ++c) {
        float4 wv = w4[c], bv = b4[c], r;
        r.x = fmaxf(fmaf(v, wv.x, bv.x), 0.f);
        r.y = fmaxf(fmaf(v, wv.y, bv.y), 0.f);
        r.z = fmaxf(fmaf(v, wv.z, bv.z), 0.f);
        r.w = fmaxf(fmaf(v, wv.w, bv.w), 0.f);
        out[c] = r;
    }
}

// ---------------------------------------------------------------------------
// xn = h / max(||h||, eps); init smax = enc(-inf), ssum = 0, hout = 0
// ---------------------------------------------------------------------------
__global__ void normalize_init_kernel(const float* __restrict__ h,
                                      float* __restrict__ xn,
                                      unsigned* __restrict__ smax_enc,
                                      float* __restrict__ ssum,
                                      float* __restrict__ hout, int n) {
    int i = blockIdx.x * blockDim.x + threadIdx.x;
    if (i >= n) return;
    const float4* hr = (const float4*)(h + (size_t)i * D);
    float4 r[4];
    float ss = 0.f;
#pragma unroll
    for (int c = 0; c < 4; ++c) {
        r[c] = hr[c];
        ss = fmaf(r[c].x, r[c].x, ss);
        ss = fmaf(r[c].y, r[c].y, ss);
        ss = fmaf(r[c].z, r[c].z, ss);
        ss = fmaf(r[c].w, r[c].w, ss);
    }
    float inv = 1.0f / fmaxf(sqrtf(ss), 1e-12f);
    float4* xo = (float4*)(xn + (size_t)i * D);
    float4* ho = (float4*)(hout + (size_t)i * D);
    float4 z = make_float4(0.f, 0.f, 0.f, 0.f);
#pragma unroll
    for (int c = 0; c < 4; ++c) {
        float4 o;
        o.x = r[c].x * inv; o.y = r[c].y * inv;
        o.z = r[c].z * inv; o.w = r[c].w * inv;
        xo[c] = o;
        ho[c] = z;
    }
    smax_enc[i] = 0x007FFFFFu;   // monotone encoding of -inf
    ssum[i] = 0.f;
}

// ---------------------------------------------------------------------------
// Per-wave: 16 edges batched through V_WMMA_F32_16X16X4_F32.
//   A rows = xn[dst[e]], B cols = xn[src[e]], diag(D) = edge dots.
// score[e] = beta * dot; atomic segment-max via monotone int encoding.
// ---------------------------------------------------------------------------
__global__ void edge_score_wmma_kernel(const float* __restrict__ xn,
                                       const int* __restrict__ src,
                                       const int* __restrict__ dst,
                                       const float* __restrict__ betas, int layer,
                                       float* __restrict__ score,
                                       unsigned* __restrict__ smax_enc, int E) {
    const int lane = threadIdx.x & 31;
    const int waveId = blockIdx.x * (blockDim.x >> 5) + (threadIdx.x >> 5);
    const int base = waveId * 16;
    if (base >= E) return;                       // wave-uniform exit

    const int r = lane & 15;
    const int half = lane >> 4;
    int e = base + r;
    bool valid = (e < E);
    int ec = valid ? e : (E - 1);                // clamp for safe loads
    int di = dst[ec];
    int si = src[ec];
    const float* drow = xn + (size_t)di * D;
    const float* srow = xn + (size_t)si * D;

    v8f c = {};
#pragma unroll
    for (int s = 0; s < 4; ++s) {
        int ka = 4 * s + (half ? 2 : 0);
        v2f a, b;
        float2 av = *(const float2*)(drow + ka);
        a.x = av.x; a.y = av.y;
        b.x = srow[4 * s + (half ? 1 : 0)];
        b.y = srow[4 * s + (half ? 3 : 2)];
        c = __builtin_amdgcn_wmma_f32_16x16x4_f32(false, a, false, b,
                                                  (short)0, c, false, false);
    }

    // diag(D): edge e<8 -> vgpr e lane e; e>=8 -> vgpr e-8 lane e+16.
    int srcLane = (lane < 8) ? lane : (lane + 16);
    float dotv = 0.f;
#pragma unroll
    for (int j = 0; j < 8; ++j) {
        float t = __shfl(c[j], srcLane, 32);
        if ((lane & 7) == j) dotv = t;
    }

    if (lane < 16 && valid) {
        float sc = betas[layer] * dotv;
        score[e] = sc;
        unsigned bits = __float_as_uint(sc);
        unsigned key = ((int)bits < 0) ? ~bits : (bits | 0x80000000u);
        atomicMax(&smax_enc[di], key);
    }
}

// ---------------------------------------------------------------------------
// ex = exp(score - smax[dst]); score <- ex; atomic ssum[dst] += ex
// ---------------------------------------------------------------------------
__global__ void edge_exp_kernel(float* __restrict__ score,
                                const unsigned* __restrict__ smax_enc,
                                const int* __restrict__ dst,
                                float* __restrict__ ssum, int E) {
    int e = blockIdx.x * blockDim.x + threadIdx.x;
    if (e >= E) return;
    int di = dst[e];
    unsigned u = smax_enc[di];
    unsigned bits = ((int)u < 0) ? (u & 0x7fffffffu) : ~u;   // decode
    float m = __uint_as_float(bits);
    float ex = __expf(score[e] - m);
    score[e] = ex;
    atomicAdd(&ssum[di], ex);
}

#ifdef USE_TDM
// ---------------------------------------------------------------------------
// TDM: DMA `tile_elems` dwords from gptr into LDS at lds_ptr.
// 1-row 2D descriptor: tensor_dim0 = remaining dwords (OOB reads -> 0),
// tile_dim0 = tile_elems, tensor_dim1 = tile_dim1 = 1. Groups 2/3 zero (<=2D).
// ---------------------------------------------------------------------------
__device__ __forceinline__ void tdm_load_dwords(void* lds_ptr, const void* gptr,
                                                unsigned remaining,
                                                unsigned tile_elems) {
    unsigned lds_off = (unsigned)(size_t)lds_ptr;   // flat LDS aperture: offset in [31:0]
    unsigned long long ga = (unsigned long long)(size_t)gptr;
    v4u g0;
    g0.x = 1u;                                          // count=1, user descriptor
    g0.y = lds_off;                                     // lds_addr [63:32]
    g0.z = (unsigned)(ga & 0xFFFFFFFFu);                // global_addr [95:64]
    g0.w = (unsigned)((ga >> 32) & 0x1FFFFFFu) | (2u << 30);  // [120:96] + type=2
    v8i_ g1;
    g1[0] = 0x00020000;                                 // data_size = 4B
    g1[1] = (int)((remaining & 0xFFFFu) << 16);         // tensor_dim0[15:0] @ bits63:48
    g1[2] = (int)(((remaining >> 16) & 0xFFFFu) | (1u << 16)); // dim0[31:16], dim1=1
    g1[3] = (int)(tile_elems << 16);                    // tile_dim0 @ bits127:112
    g1[4] = 1;                                          // tile_dim1 = 1
    g1[5] = (int)remaining;                             // dim0 stride (single row)
    g1[6] = 0;
    g1[7] = 0;
    v4i_ gz = {0, 0, 0, 0};
#if defined(__clang_major__) && __clang_major__ >= 23
    v8i_ gz8 = {0, 0, 0, 0, 0, 0, 0, 0};
    __builtin_amdgcn_tensor_load_to_lds(g0, g1, gz, gz, gz8, 0);
#else
    __builtin_amdgcn_tensor_load_to_lds(g0, g1, gz, gz, 0);
#endif
}
#endif

// ---------------------------------------------------------------------------
// Block owns AGG_CHUNK edges: stage src/dst/ex chunk into LDS (TDM DMA when
// available), then half-wave (16 lanes) per edge: coalesced 64B row gather of
// hin[src], 16 contiguous L2-resident float atomics into hout[dst].
// ---------------------------------------------------------------------------
__global__ void aggregate_kernel(const float* __restrict__ hin,
                                 const float* __restrict__ ex,
                                 const float* __restrict__ ssum,
                                 const int* __restrict__ src,
                                 const int* __restrict__ dst,
                                 float* __restrict__ hout, int E) {
    __shared__ int   lsrc[AGG_CHUNK];
    __shared__ int   ldst[AGG_CHUNK];
    __shared__ float lex[AGG_CHUNK];

    const int base = blockIdx.x * AGG_CHUNK;
    const int tid = threadIdx.x;

#ifdef USE_TDM
    if (tid < 32) {                     // wave 0 issues the three DMA descriptors
        unsigned rem = (unsigned)(E - base);
        tdm_load_dwords(lsrc, src + base, rem, AGG_CHUNK);
        tdm_load_dwords(ldst, dst + base, rem, AGG_CHUNK);
        tdm_load_dwords(lex,  ex  + base, rem, AGG_CHUNK);
        __builtin_amdgcn_s_wait_tensorcnt(0);
    }
#else
    for (int i = tid; i < AGG_CHUNK; i += 256) {
        int e = base + i;
        lsrc[i] = (e < E) ? src[e] : 0;
        ldst[i] = (e < E) ? dst[e] : 0;
        lex[i]  = (e < E) ? ex[e]  : 0.f;
    }
#endif
    __syncthreads();

    const int hw = tid >> 4;            // half-wave id: 0..15
    const int d  = tid & 15;            // feature lane
    for (int i = hw; i < AGG_CHUNK; i += 16) {
        int e = base + i;
        if (e >= E) break;
        if (d == 0 && (e + 16) < E)     // one lookahead prefetch per edge
            __builtin_prefetch(&hin[(size_t)lsrc[i + 16] * D], 0, 1);
        int di = ldst[i];
        int si = lsrc[i];
        float alpha = lex[i] / ssum[di];
        atomicAdd(&hout[(size_t)di * D + d], alpha * hin[(size_t)si * D + d]);
    }
}

// ---------------------------------------------------------------------------
// out[n] = dot(h[n], out_w) + out_b
// ---------------------------------------------------------------------------
__global__ void out_proj_kernel(const float* __restrict__ h,
                                const float* __restrict__ w,
                                const float* __restrict__ b,
                                float* __restrict__ out, int n) {
    int i = blockIdx.x * blockDim.x + threadIdx.x;
    if (i >= n) return;
    const float4* hr = (const float4*)(h + (size_t)i * D);
    const float4* w4 = (const float4*)w;
    float acc = 0.f;
#pragma unroll
    for (int c = 0; c < 4; ++c) {
        float4 hv = hr[c], wv = w4[c];
        acc = fmaf(hv.x, wv.x, acc);
        acc = fmaf(hv.y, wv.y, acc);
        acc = fmaf(hv.z, wv.z, acc);
        acc = fmaf(hv.w, wv.w, acc);
    }
    out[i] = acc + b[0];
}

// ---------------------------------------------------------------------------
extern "C" void kernel_launch(void* const* d_in, const int* in_sizes, int n_in,
                              void* d_out, int out_size, void* d_ws, size_t ws_size,
                              hipStream_t stream) {
    const float* x      = (const float*)d_in[0];   // [N,1]
    const int*   eidx   = (const int*)  d_in[1];   // [2,E]
    const float* lin_w  = (const float*)d_in[2];   // [1,16]
    const float* lin_b  = (const float*)d_in[3];   // [16]
    const float* betas  = (const float*)d_in[4];   // [2]
    const float* out_w  = (const float*)d_in[5];   // [16,1]
    const float* out_b  = (const float*)d_in[6];   // [1]

    const int N = in_sizes[0];
    const int E = in_sizes[1] / 2;
    const int* src = eidx;          // row 0
    const int* dst = eidx + E;      // row 1

    float*    hA    = (float*)d_ws;
    float*    hB    = hA + (size_t)N * D;
    float*    xn    = hB + (size_t)N * D;
    float*    score = xn + (size_t)N * D;
    unsigned* smax  = (unsigned*)(score + E);
    float*    ssum  = (float*)(smax + N);

    const int nodeBlocks = (N + 255) / 256;
    lin_relu_kernel<<<nodeBlocks, 256, 0, stream>>>(x, lin_w, lin_b, hA, N);

    for (int layer = 0; layer < 2; ++layer) {
        float* hin  = layer ? hB : hA;
        float* hout = layer ? hA : hB;

        normalize_init_kernel<<<nodeBlocks, 256, 0, stream>>>(hin, xn, smax, ssum, hout, N);

        long long waves  = ((long long)E + 15) / 16;
        int scoreBlocks  = (int)((waves + 7) / 8);          // 8 waves / block
        edge_score_wmma_kernel<<<scoreBlocks, 256, 0, stream>>>(
            xn, src, dst, betas, layer, score, smax, E);

        int expBlocks = (E + 255) / 256;
        edge_exp_kernel<<<expBlocks, 256, 0, stream>>>(score, smax, dst, ssum, E);

        int aggBlocks = (E + AGG_CHUNK - 1) / AGG_CHUNK;
        aggregate_kernel<<<aggBlocks, 256, 0, stream>>>(hin, score, ssum, src, dst, hout, E);
    }

    out_proj_kernel<<<nodeBlocks, 256, 0, stream>>>(hA, out_w, out_b, (float*)d_out, N);
}